// VectorQuant_4406636446030
// MI455X (gfx1250) — compile-verified
//
#include <hip/hip_runtime.h>
#include <hip/hip_bf16.h>
#include <math.h>

typedef __attribute__((ext_vector_type(2))) float v2f;
typedef __attribute__((ext_vector_type(8))) float v8f;

// Problem constants: N=8, S=4096, C=2, V=128, K=512
#define TOKENS   32768      // N*S
#define CH       2
#define VDIM     128
#define KCODE    512
#define OUT1_OFF 8388608    // TOKENS*CH*VDIM
#define OUT2_OFF 8454144    // OUT1_OFF + TOKENS*CH
#define ENT_OFF  8519680    // OUT2_OFF + TOKENS*CH

// ---------------------------------------------------------------------------
// Kernel 0: zero the histogram, precompute ||e_k||^2 per (c,k)
// ---------------------------------------------------------------------------
__global__ __launch_bounds__(256) void vq_setup(const float* __restrict__ emb,
                                                unsigned int* __restrict__ hist,
                                                float* __restrict__ esq) {
    int tid = blockIdx.x * 256 + threadIdx.x;
    if (tid < KCODE) hist[tid] = 0u;
    if (tid < CH * KCODE) {
        const float* ep = emb + (size_t)tid * VDIM;
        float s = 0.0f;
        #pragma unroll 8
        for (int v = 0; v < VDIM; ++v) { float e = ep[v]; s += e * e; }
        esq[tid] = s;
    }
}

// ---------------------------------------------------------------------------
// Kernel 1: FP32-WMMA nearest-neighbor search + gather + residuals + histogram
//   grid = (TOKENS/256)*CH blocks of 256 threads (8 wave32).
//   Each wave: 32 tokens (two 16-row A tiles kept in VGPRs), all 512 codes.
// ---------------------------------------------------------------------------
__global__ __launch_bounds__(256) void vq_main(const float* __restrict__ x0,
                                               const float* __restrict__ emb,
                                               const float* __restrict__ esq,
                                               unsigned int* __restrict__ hist,
                                               float* __restrict__ out) {
    __shared__ int sIdx[256];

    const int lane   = threadIdx.x & 31;
    const int waveId = threadIdx.x >> 5;
    const int lane16 = lane & 15;
    const int half   = lane >> 4;          // 0: K=0,1 / cols 0-15 ; 1: K=2,3
    const int c       = blockIdx.x & 1;
    const int rowBase = (blockIdx.x >> 1) * 256;
    const int waveRow = rowBase + waveId * 32;

    // --- Load A fragments (ISA 16x4 FP32 A layout), 2 row-tiles resident ---
    v2f a0[32], a1[32];
    {
        const float* xp0 = x0 + (size_t)(waveRow + lane16) * (CH * VDIM)
                              + c * VDIM + half * 2;
        const float* xp1 = xp0 + 16 * (CH * VDIM);
        #pragma unroll
        for (int kk = 0; kk < 32; ++kk) {
            a0[kk] = *(const v2f*)(xp0 + kk * 4);
            a1[kk] = *(const v2f*)(xp1 + kk * 4);
        }
    }

    float runD0[8], runD1[8];
    int   runI0[8], runI1[8];
    #pragma unroll
    for (int r = 0; r < 8; ++r) {
        runD0[r] = 3.4e38f; runD1[r] = 3.4e38f; runI0[r] = 0; runI1[r] = 0;
    }

    // --- Sweep all 32 column tiles (16 codes each) of this channel ---
    for (int t = 0; t < 32; ++t) {
        const int code = t * 16 + lane16;             // this lane's column code
        const float ev2 = esq[c * KCODE + code];
        const float* ep = emb + ((size_t)(c * KCODE + code)) * VDIM + half * 2;

        v8f acc0 = {}; v8f acc1 = {};
        #pragma unroll
        for (int kk = 0; kk < 32; ++kk) {
            v2f b = *(const v2f*)(ep + kk * 4);       // B frag (4x16 KxN layout)
            acc0 = __builtin_amdgcn_wmma_f32_16x16x4_f32(
                false, a0[kk], false, b, (short)0, acc0, false, false);
            acc1 = __builtin_amdgcn_wmma_f32_16x16x4_f32(
                false, a1[kk], false, b, (short)0, acc1, false, false);
        }
        // dist2 = ||e||^2 - 2*dot ; keep running (min, argmin) per lane/row
        #pragma unroll
        for (int r = 0; r < 8; ++r) {
            float d0 = ev2 - 2.0f * acc0[r];
            float d1 = ev2 - 2.0f * acc1[r];
            if (d0 < runD0[r]) { runD0[r] = d0; runI0[r] = code; }
            if (d1 < runD1[r]) { runD1[r] = d1; runI1[r] = code; }
        }
    }

    // --- Butterfly min over the 16 columns (stays within 16-lane halves) ---
    #pragma unroll
    for (int m = 1; m < 16; m <<= 1) {
        #pragma unroll
        for (int r = 0; r < 8; ++r) {
            float od = __shfl_xor(runD0[r], m, 32);
            int   oi = __shfl_xor(runI0[r], m, 32);
            if (od < runD0[r] || (od == runD0[r] && oi < runI0[r])) {
                runD0[r] = od; runI0[r] = oi;
            }
            od = __shfl_xor(runD1[r], m, 32);
            oi = __shfl_xor(runI1[r], m, 32);
            if (od < runD1[r] || (od == runD1[r] && oi < runI1[r])) {
                runD1[r] = od; runI1[r] = oi;
            }
        }
    }

    // --- Publish indices (rows M = r + 8*half per C/D VGPR layout) ---------
    if (lane16 == 0) {
        #pragma unroll
        for (int r = 0; r < 8; ++r) {
            int rl0 = waveId * 32 + r + 8 * half;
            sIdx[rl0] = runI0[r];
            atomicAdd(&hist[runI0[r]], 1u);
            int rl1 = rl0 + 16;
            sIdx[rl1] = runI1[r];
            atomicAdd(&hist[runI1[r]], 1u);
        }
    }
    __syncthreads();

    // --- Gather phase: out0 = e[idx], out1 = out2 = ||x - e[idx]||^2 -------
    // One wave per row per iteration; x re-read hits L2 (whole tensor fits).
    for (int it = 0; it < 32; ++it) {
        int rloc = waveId + it * 8;
        int row  = rowBase + rloc;
        int idx  = sIdx[rloc];
        const float4 ev = *(const float4*)(emb + ((size_t)(c * KCODE + idx)) * VDIM + lane * 4);
        const float4 xv = *(const float4*)(x0 + (size_t)row * (CH * VDIM) + c * VDIM + lane * 4);
        *(float4*)(out + (size_t)row * (CH * VDIM) + c * VDIM + lane * 4) = ev;
        float dx = xv.x - ev.x, dy = xv.y - ev.y;
        float dz = xv.z - ev.z, dw = xv.w - ev.w;
        float s = dx * dx + dy * dy + dz * dz + dw * dw;
        #pragma unroll
        for (int m = 1; m < 32; m <<= 1) s += __shfl_xor(s, m, 32);
        if (lane == 0) {
            out[OUT1_OFF + row * CH + c] = s;
            out[OUT2_OFF + row * CH + c] = s;   // (x - x0) == 0 in this path
        }
    }
}

// ---------------------------------------------------------------------------
// Kernel 2: entropy of codebook usage (reference divides counts by B=32768)
// ---------------------------------------------------------------------------
__global__ __launch_bounds__(256) void vq_entropy(const unsigned int* __restrict__ hist,
                                                  float* __restrict__ out) {
    __shared__ float red[256];
    int t = threadIdx.x;
    float s = 0.0f;
    for (int k = t; k < KCODE; k += 256) {
        unsigned int h = hist[k];
        if (h > 0u) {
            float p = (float)h * (1.0f / 32768.0f);
            s += p * logf(p);
        }
    }
    red[t] = s;
    __syncthreads();
    for (int o = 128; o > 0; o >>= 1) {
        if (t < o) red[t] += red[t + o];
        __syncthreads();
    }
    if (t == 0) out[ENT_OFF] = -red[0];
}

// ---------------------------------------------------------------------------
extern "C" void kernel_launch(void* const* d_in, const int* in_sizes, int n_in,
                              void* d_out, int out_size, void* d_ws, size_t ws_size,
                              hipStream_t stream) {
    const float* x0  = (const float*)d_in[0];   // (8,4096,2,128) f32
    const float* emb = (const float*)d_in[1];   // (2,512,128)    f32
    float* out = (float*)d_out;

    unsigned int* hist = (unsigned int*)d_ws;                         // 512 u32
    float* esq = (float*)((char*)d_ws + KCODE * sizeof(unsigned int)); // 1024 f32

    vq_setup  <<<4, 256, 0, stream>>>(emb, hist, esq);
    vq_main   <<<(TOKENS / 256) * CH, 256, 0, stream>>>(x0, emb, esq, hist, out);
    vq_entropy<<<1, 256, 0, stream>>>(hist, out);
}